// LIFNeuronLayer_87557203296887
// MI455X (gfx1250) — compile-verified
//
#include <hip/hip_runtime.h>
#include <cstdint>

// LIF scan, fp32, bandwidth-bound (~402 MB => ~17us floor @ 23.3 TB/s).
// Per-wave async global->LDS pipeline, 64 time steps deep (ASYNCcnt),
// b128 async loads covering 4 time-step rows per instruction, no barriers
// (each wave loads and consumes only its own 32-lane LDS segment),
// non-temporal streaming stores so outputs don't evict the L2 input stream.

#define LIF_BETA 0.904837f
#define HDIM   2048
#define BLOCK  256
#define TGROUP 4                       // time steps per async b128 group
#define GDEPTH 16                      // groups in flight  (wait imm = 15)
#define STAGES (TGROUP * GDEPTH)       // 64 stages * 256 lanes * 4B = 64 KiB LDS

__global__ __launch_bounds__(BLOCK) void lif_scan_kernel(
    const float* __restrict__ in,      // (B, T, H)
    const float* __restrict__ v0,      // (B, H)
    float* __restrict__ spikes,        // (B, T, H)
    float* __restrict__ membrane,      // (B, T, H)
    float* __restrict__ vfinal,        // (B, H)
    int T)                             // T % TGROUP == 0, T >= STAGES
{
  __shared__ float buf[STAGES][BLOCK];

  const int tid = threadIdx.x;
  const int w   = tid >> 5;            // wave id within block (wave32)
  const int l   = tid & 31;            // lane id
  const int dt  = l >> 3;              // which of the 4 time rows this lane fetches
  const int q   = l & 7;               // which 16B chunk of the wave's 128B row seg

  const int hchunks = HDIM / BLOCK;    // 8
  const int b  = blockIdx.x / hchunks;
  const int hb = (blockIdx.x % hchunks) * BLOCK;
  const int h  = hb + tid;

  const size_t rowbase = ((size_t)b * T) * HDIM;

  // per-lane async source pointer for group 0: row (t=dt), chunk q of wave w's segment
  const float* gl = in + rowbase + hb + 32 * w + (size_t)dt * HDIM + 4 * q;
  // per-lane LDS destination byte offset for group-slot 0
  const uint32_t lbase = (uint32_t)(uintptr_t)&buf[0][0]
                       + (uint32_t)(dt * (BLOCK * 4) + 128 * w + 16 * q);

  // ---- prologue: fill all GDEPTH group slots (time steps 0..STAGES-1) ----
#pragma unroll
  for (int gs = 0; gs < GDEPTH; ++gs) {
    asm volatile("global_load_async_to_lds_b128 %0, %1, off"
                 :: "v"(lbase + gs * (TGROUP * BLOCK * 4)),
                    "v"((uint64_t)(uintptr_t)(gl + (size_t)gs * TGROUP * HDIM))
                 : "memory");
  }
  const float* pref = gl + (size_t)STAGES * HDIM;

  float v    = v0[(size_t)b * HDIM + h];
  float keep = LIF_BETA;               // beta * (1 - s_prev); s_prev = 0

  float* sp = spikes   + rowbase + h;
  float* mb = membrane + rowbase + h;

  const int NG = T / TGROUP;

  // ---- steady state: one wait + one b128 async per 4 time steps ----
  int g = 0;
  for (; g < NG - GDEPTH; ++g) {
    // GDEPTH groups outstanding; <=15 remaining => oldest (group g) complete
    asm volatile("s_wait_asynccnt 15" ::: "memory");
    const int gs = g & (GDEPTH - 1);
    const float* lrow = &buf[gs * TGROUP][tid];
#pragma unroll
    for (int k = 0; k < TGROUP; ++k) {
      float it = lrow[(size_t)k * BLOCK];           // ds_load, imm offset k*1024
      v = __builtin_fmaf(keep, v, it);              // v = beta*(1-s_prev)*v + i_t
      float s = (v >= 1.0f) ? 1.0f : 0.0f;
      keep    = (v >= 1.0f) ? 0.0f : LIF_BETA;
      __builtin_nontemporal_store(s, sp + (size_t)k * HDIM);  // imm offset k*8192
      __builtin_nontemporal_store(v, mb + (size_t)k * HDIM);
    }
    sp += (size_t)TGROUP * HDIM;
    mb += (size_t)TGROUP * HDIM;

    // refill the slot just consumed with group g+GDEPTH.
    // Safe: all 4 ds_load results were consumed by VALU above (DScnt waited),
    // and the volatile asm w/ memory clobber orders all memory ops.
    asm volatile("global_load_async_to_lds_b128 %0, %1, off"
                 :: "v"(lbase + gs * (TGROUP * BLOCK * 4)),
                    "v"((uint64_t)(uintptr_t)pref)
                 : "memory");
    pref += (size_t)TGROUP * HDIM;
  }

  // ---- drain tail: all remaining groups already issued ----
  asm volatile("s_wait_asynccnt 0" ::: "memory");
  for (; g < NG; ++g) {
    const int gs = g & (GDEPTH - 1);
    const float* lrow = &buf[gs * TGROUP][tid];
#pragma unroll
    for (int k = 0; k < TGROUP; ++k) {
      float it = lrow[(size_t)k * BLOCK];
      v = __builtin_fmaf(keep, v, it);
      float s = (v >= 1.0f) ? 1.0f : 0.0f;
      keep    = (v >= 1.0f) ? 0.0f : LIF_BETA;
      __builtin_nontemporal_store(s, sp + (size_t)k * HDIM);
      __builtin_nontemporal_store(v, mb + (size_t)k * HDIM);
    }
    sp += (size_t)TGROUP * HDIM;
    mb += (size_t)TGROUP * HDIM;
  }

  vfinal[(size_t)b * HDIM + h] = v;
}

extern "C" void kernel_launch(void* const* d_in, const int* in_sizes, int n_in,
                              void* d_out, int out_size, void* d_ws, size_t ws_size,
                              hipStream_t stream) {
  const float* in = (const float*)d_in[0];   // input_current (B,T,H)
  const float* v0 = (const float*)d_in[1];   // v_init (B,H)

  const int BH = in_sizes[1];                // B*H
  const int B  = BH / HDIM;
  const int T  = in_sizes[0] / BH;           // 1024 (mult. of TGROUP, >= STAGES)

  float* out      = (float*)d_out;
  const size_t n  = (size_t)in_sizes[0];     // B*T*H
  float* spikes   = out;                     // output 0
  float* membrane = out + n;                 // output 1
  float* vfinal   = out + 2 * n;             // output 2

  dim3 grid(B * (HDIM / BLOCK));             // 16 * 8 = 128 blocks, 1024 waves
  lif_scan_kernel<<<grid, BLOCK, 0, stream>>>(in, v0, spikes, membrane, vfinal, T);
}